// GraphUNet_25580825215333
// MI455X (gfx1250) — compile-verified
//
#include <hip/hip_runtime.h>

typedef float v2f __attribute__((ext_vector_type(2)));
typedef float v8f __attribute__((ext_vector_type(8)));

// ---------------------------------------------------------------------------
// Fused dense / SAGE matmul:  Y = act( (M ? M@Wl : 0) + X@Wr + bias )
// One 16x16 output tile row per wave32 across NT column tiles;
// K-loop of V_WMMA_F32_16X16X4_F32 with A loaded once per K-step and reused
// across all NT column tiles. Weights zero-padded to CoutP columns in LDS so
// the inner loop has no predication (EXEC stays all-ones around WMMA).
// ---------------------------------------------------------------------------
template <int COUT>
__global__ __launch_bounds__(128) void wmma_linear_t(
    const float* __restrict__ X, const float* __restrict__ Wr,
    const float* __restrict__ M, const float* __restrict__ Wl,
    const float* __restrict__ bias, float* __restrict__ Y,
    int N, int Cin, int relu) {
  constexpr int NT = (COUT + 15) / 16;   // column tiles
  constexpr int CoutP = NT * 16;         // padded columns
  extern __shared__ float lds[];
  float* sWr = lds;                                  // Cin * CoutP
  float* sWl = lds + Cin * CoutP;                    // Cin * CoutP (if fused)
  float* sB  = lds + (M ? 2 : 1) * (size_t)Cin * CoutP;  // CoutP

  const int wsz = Cin * CoutP;
  for (int t = threadIdx.x; t < wsz; t += blockDim.x) {
    const int k = t / CoutP, c = t - k * CoutP;
    sWr[t] = (c < COUT) ? Wr[k * COUT + c] : 0.f;
  }
  if (M != nullptr)
    for (int t = threadIdx.x; t < wsz; t += blockDim.x) {
      const int k = t / CoutP, c = t - k * CoutP;
      sWl[t] = (c < COUT) ? Wl[k * COUT + c] : 0.f;
    }
  for (int t = threadIdx.x; t < CoutP; t += blockDim.x)
    sB[t] = (t < COUT) ? bias[t] : 0.f;
  __syncthreads();

  const int lane  = threadIdx.x & 31;
  const int wave  = threadIdx.x >> 5;
  const int m0    = (blockIdx.x * 4 + wave) * 16;
  if (m0 >= N) return;  // wave-uniform: EXEC all-ones for WMMA below

  const int nlane = lane & 15;
  const int hi    = lane >> 4;       // 0: K-pair {0,1}; 1: K-pair {2,3}
  const int koff  = hi * 2;
  int row = m0 + nlane;
  if (row >= N) row = N - 1;         // clamped load; stores guarded below

  v8f acc[NT];
#pragma unroll
  for (int t = 0; t < NT; ++t) acc[t] = v8f{0.f,0.f,0.f,0.f,0.f,0.f,0.f,0.f};

  {
    const float* xr   = X + (size_t)row * Cin + koff;
    const float* bcol = sWr + koff * CoutP + nlane;
#pragma unroll 2
    for (int k = 0; k < Cin; k += 4) {
      const v2f av = *reinterpret_cast<const v2f*>(xr + k);
      const float* bk = bcol + k * CoutP;
#pragma unroll
      for (int t = 0; t < NT; ++t) {
        v2f bv; bv.x = bk[t * 16]; bv.y = bk[t * 16 + CoutP];
        acc[t] = __builtin_amdgcn_wmma_f32_16x16x4_f32(
            false, av, false, bv, (short)0, acc[t], false, false);
      }
    }
  }
  if (M != nullptr) {
    const float* mr   = M + (size_t)row * Cin + koff;
    const float* bcol = sWl + koff * CoutP + nlane;
#pragma unroll 2
    for (int k = 0; k < Cin; k += 4) {
      const v2f av = *reinterpret_cast<const v2f*>(mr + k);
      const float* bk = bcol + k * CoutP;
#pragma unroll
      for (int t = 0; t < NT; ++t) {
        v2f bv; bv.x = bk[t * 16]; bv.y = bk[t * 16 + CoutP];
        acc[t] = __builtin_amdgcn_wmma_f32_16x16x4_f32(
            false, av, false, bv, (short)0, acc[t], false, false);
      }
    }
  }

  const bool fullM = (m0 + 16 <= N);
#pragma unroll
  for (int t = 0; t < NT; ++t) {
    const int c = t * 16 + nlane;
    const bool cv = (COUT % 16 == 0) ? true : (c < COUT);
    const float bb = sB[c];
#pragma unroll
    for (int r = 0; r < 8; ++r) {
      const int orow = m0 + r + hi * 8;   // C/D layout: VGPR r, halves at M+8
      float v = acc[t][r] + bb;
      if (relu) v = fmaxf(v, 0.f);
      if (cv && (fullM || orow < N)) Y[(size_t)orow * COUT + c] = v;
    }
  }
}

// ---------------------------------------------------------------------------
// Mean aggregation: every node has exactly 16 in-edges (dst = repeat(arange)).
// ---------------------------------------------------------------------------
__global__ __launch_bounds__(256) void sage_agg(
    const float* __restrict__ X, const int* __restrict__ src,
    float* __restrict__ Mo, int N, int C4) {
  const int idx = blockIdx.x * blockDim.x + threadIdx.x;
  if (idx >= N * C4) return;
  const int i = idx / C4;
  const int c = (idx - i * C4) * 4;
  const int C = C4 * 4;
  const int* e = src + (size_t)i * 16;
  float4 s = make_float4(0.f, 0.f, 0.f, 0.f);
#pragma unroll
  for (int k = 0; k < 16; ++k) {
    const int j = e[k];
    const float4 v = *reinterpret_cast<const float4*>(X + (size_t)j * C + c);
    s.x += v.x; s.y += v.y; s.z += v.z; s.w += v.w;
  }
  const float4 o = make_float4(s.x * 0.0625f, s.y * 0.0625f,
                               s.z * 0.0625f, s.w * 0.0625f);
  *reinterpret_cast<float4*>(Mo + (size_t)i * C + c) = o;
}

// Y[i][:] = X[ids[i]][:]
__global__ __launch_bounds__(256) void row_gather(
    const float* __restrict__ X, const int* __restrict__ ids,
    float* __restrict__ Y, int N, int C4) {
  const int idx = blockIdx.x * blockDim.x + threadIdx.x;
  if (idx >= N * C4) return;
  const int i = idx / C4;
  const int c = (idx - i * C4) * 4;
  const int C = C4 * 4;
  *reinterpret_cast<float4*>(Y + (size_t)i * C + c) =
      *reinterpret_cast<const float4*>(X + (size_t)ids[i] * C + c);
}

// Y[i][0:Cz] = Z[cluster[i]];  Y[i][Cz:Cz+Cs] = S[i]
__global__ __launch_bounds__(256) void up_concat(
    const float* __restrict__ Z, const int* __restrict__ cl,
    const float* __restrict__ S, float* __restrict__ Y,
    int N, int Cz, int Cs) {
  const int C = Cz + Cs;
  const int C4 = C / 4;
  const int idx = blockIdx.x * blockDim.x + threadIdx.x;
  if (idx >= N * C4) return;
  const int i = idx / C4;
  const int c = (idx - i * C4) * 4;
  float4 v;
  if (c < Cz)
    v = *reinterpret_cast<const float4*>(Z + (size_t)cl[i] * Cz + c);
  else
    v = *reinterpret_cast<const float4*>(S + (size_t)i * Cs + (c - Cz));
  *reinterpret_cast<float4*>(Y + (size_t)i * C + c) = v;
}

// ---------------------------------------------------------------------------
// Host orchestration
// ---------------------------------------------------------------------------
static inline int cdiv(int a, int b) { return (a + b - 1) / b; }

template <int COUT>
static void launch_linear_t(const float* X, const float* Wr, const float* M,
                            const float* Wl, const float* b, float* Y,
                            int N, int Cin, int relu, hipStream_t s) {
  constexpr int CoutP = ((COUT + 15) / 16) * 16;
  const size_t lds = ((size_t)Cin * CoutP * (M ? 2 : 1) + CoutP) * sizeof(float);
  wmma_linear_t<COUT><<<cdiv(N, 64), 128, lds, s>>>(X, Wr, M, Wl, b, Y, N, Cin, relu);
}

static void launch_linear(const float* X, const float* Wr, const float* M,
                          const float* Wl, const float* b, float* Y,
                          int N, int Cin, int Cout, int relu, hipStream_t s) {
  switch (Cout) {
    case 4:   launch_linear_t<4>(X, Wr, M, Wl, b, Y, N, Cin, relu, s); break;
    case 8:   launch_linear_t<8>(X, Wr, M, Wl, b, Y, N, Cin, relu, s); break;
    case 16:  launch_linear_t<16>(X, Wr, M, Wl, b, Y, N, Cin, relu, s); break;
    case 32:  launch_linear_t<32>(X, Wr, M, Wl, b, Y, N, Cin, relu, s); break;
    case 64:  launch_linear_t<64>(X, Wr, M, Wl, b, Y, N, Cin, relu, s); break;
    case 128: launch_linear_t<128>(X, Wr, M, Wl, b, Y, N, Cin, relu, s); break;
    default: break;  // not reached for this network
  }
}

extern "C" void kernel_launch(void* const* d_in, const int* in_sizes, int n_in,
                              void* d_out, int out_size, void* d_ws, size_t ws_size,
                              hipStream_t stream) {
  (void)n_in; (void)out_size; (void)ws_size;
  auto F = [&](int i) { return (const float*)d_in[i]; };
  auto I = [&](int i) { return (const int*)d_in[i]; };

  // Input layout (setup_inputs dict order, lists flattened):
  // 0:a 1:geom 2-4:enc_Ws 5-7:enc_bs 8-10:dec_Ws 11-13:dec_bs
  // 14-18:down_Wl 19-23:down_Wr 24-28:down_b 29-32:up_Wl 33-36:up_Wr 37-40:up_b
  // 41-45:edges_src 46-50:edges_dst 51-54:ids 55-58:clusters 59-63:num_nodes
  const float* a = F(0);
  const float *encW[3], *encB[3], *decW[3], *decB[3];
  for (int i = 0; i < 3; ++i) {
    encW[i] = F(2 + i); encB[i] = F(5 + i);
    decW[i] = F(8 + i); decB[i] = F(11 + i);
  }
  const float *dWl[5], *dWr[5], *dB[5];
  for (int i = 0; i < 5; ++i) { dWl[i] = F(14 + i); dWr[i] = F(19 + i); dB[i] = F(24 + i); }
  const float *uWl[4], *uWr[4], *uB[4];
  for (int i = 0; i < 4; ++i) { uWl[i] = F(29 + i); uWr[i] = F(33 + i); uB[i] = F(37 + i); }
  const int* esrc[5]; for (int i = 0; i < 5; ++i) esrc[i] = I(41 + i);
  const int* ids[4];  for (int i = 0; i < 4; ++i) ids[i]  = I(51 + i);
  const int* clus[4]; for (int i = 0; i < 4; ++i) clus[i] = I(55 + i);

  int ns[5];
  ns[0] = in_sizes[0] / 8;                      // a is N0 x 8
  for (int l = 1; l <= 4; ++l) ns[l] = in_sizes[51 + (l - 1)];  // ids sizes

  const int CinD[5]  = {8, 8, 16, 32, 64};
  const int CoutD[5] = {8, 16, 32, 64, 128};
  const int CzU[4] = {16, 32, 64, 128};         // incoming up-feature channels
  const int CsU[4] = {8, 16, 32, 64};           // skip channels (z_l)
  const int CoU[4] = {8, 16, 32, 64};           // up SAGE output channels

  // Workspace carving (floats), 16B-aligned via 64-float rounding.
  float* w = (float*)d_ws;
  size_t off = 0;
  auto alloc = [&](size_t nf) { float* p = w + off; off += (nf + 63) & ~(size_t)63; return p; };
  float* z[5];
  for (int l = 0; l <= 4; ++l) z[l] = alloc((size_t)ns[l] * CoutD[l]);
  size_t sBig = (size_t)ns[3] * 192;                       // largest concat input
  size_t encH = (size_t)ns[0] * 64;                        // enc/dec hidden
  if (encH > sBig) sBig = encH;
  float* s0 = alloc(sBig);                                 // X / concat scratch
  float* s1 = alloc(sBig);                                 // mean-agg scratch
  size_t sUp = (size_t)ns[3] * 64;                         // largest up output
  float* s2 = alloc(sUp);
  float* s3 = alloc(sUp);

  // ---------------- Encoder MLP: 8 -> 64 -> 64 -> 8 ----------------
  launch_linear(a,  encW[0], nullptr, nullptr, encB[0], s0, ns[0], 8,  64, 1, stream);
  launch_linear(s0, encW[1], nullptr, nullptr, encB[1], s1, ns[0], 64, 64, 1, stream);
  launch_linear(s1, encW[2], nullptr, nullptr, encB[2], s2, ns[0], 64, 8,  0, stream);

  // ---------------- SAGE level 0 (8 -> 8) ----------------
  sage_agg<<<cdiv(ns[0] * 2, 256), 256, 0, stream>>>(s2, esrc[0], s0, ns[0], 2);
  launch_linear(s2, dWr[0], s0, dWl[0], dB[0], z[0], ns[0], 8, 8, 1, stream);

  // ---------------- Down path ----------------
  for (int n = 0; n < 4; ++n) {
    const int Cin = CinD[n + 1], Cout = CoutD[n + 1], Nn = ns[n + 1];
    const int C4 = Cin / 4;
    row_gather<<<cdiv(Nn * C4, 256), 256, 0, stream>>>(z[n], ids[n], s0, Nn, C4);
    sage_agg<<<cdiv(Nn * C4, 256), 256, 0, stream>>>(s0, esrc[n + 1], s1, Nn, C4);
    launch_linear(s0, dWr[n + 1], s1, dWl[n + 1], dB[n + 1], z[n + 1],
                  Nn, Cin, Cout, 1, stream);
  }

  // ---------------- Up path ----------------
  float* zc = z[4];
  float* outs[4] = {s3, s2, s3, s2};  // l=0 -> s3 (final), 1->s2, 2->s3, 3->s2
  for (int l = 3; l >= 0; --l) {
    const int Cz = CzU[l], Cs = CsU[l], Cin = Cz + Cs, Cout = CoU[l], Nl = ns[l];
    const int C4 = Cin / 4;
    up_concat<<<cdiv(Nl * C4, 256), 256, 0, stream>>>(zc, clus[l], z[l], s0, Nl, Cz, Cs);
    sage_agg<<<cdiv(Nl * C4, 256), 256, 0, stream>>>(s0, esrc[l], s1, Nl, C4);
    launch_linear(s0, uWr[l], s1, uWl[l], uB[l], outs[l], Nl, Cin, Cout, 1, stream);
    zc = outs[l];
  }

  // ---------------- Decoder MLP: 8 -> 64 -> 64 -> 4 ----------------
  launch_linear(zc, decW[0], nullptr, nullptr, decB[0], s0, ns[0], 8,  64, 1, stream);
  launch_linear(s0, decW[1], nullptr, nullptr, decB[1], s1, ns[0], 64, 64, 1, stream);
  launch_linear(s1, decW[2], nullptr, nullptr, decB[2], (float*)d_out,
                ns[0], 64, 4, 0, stream);
}